// FastLoraContextRefinerBlock_27195732918670
// MI455X (gfx1250) — compile-verified
//
#include <hip/hip_runtime.h>
#include <cmath>

typedef __attribute__((ext_vector_type(2))) float v2f;
typedef __attribute__((ext_vector_type(8))) float v8f;

#define H_DIM 2048
#define FFN   8192
#define LSEQ  512
#define EPS   1e-5f

// ---- workspace layout (float element offsets) ----
#define WS_W2   0                         // 16*2048  merged ln1_w * A2
#define WS_W3   (16*2048)                 // 16*2048  merged ln1_w * A3
#define WS_C2   (2*16*2048)               // 16 (then d2,c3,d3 at +16,+32,+48)
#define WS_SS   (WS_C2 + 64)              // 64 tokens * 2 halves * 256 partials
#define WS_HP   (WS_SS + 64*512)          // 64*2048  h_pooled
#define WS_HN   (WS_HP + 64*2048)         // 64*2048  LN2(h_pooled)
#define WS_HT   (WS_HN + 64*2048)         // 64*8192  gelu(ffn1)
#define WS_P2   (WS_HT + 64*8192)         // 2 * 64*2048  ffn2 split-K partials

__device__ __forceinline__ v8f wmma4(v2f a, v2f b, v8f c) {
  // D = A(16x4 f32) * B(4x16 f32) + C(16x16 f32)
  return __builtin_amdgcn_wmma_f32_16x16x4_f32(false, a, false, b, (short)0, c,
                                               false, false);
}

// ------------------------------------------------------------------
// K0: merged weights W2 = ln1_w .* A2 (and A3), plus correction sums
//     c = sum_h ln1_w[h]*A[r,h], d = sum_h ln1_b[h]*A[r,h]
// ------------------------------------------------------------------
__global__ __launch_bounds__(256) void prep_kernel(
    const float* __restrict__ A2, const float* __restrict__ A3,
    const float* __restrict__ lw, const float* __restrict__ lb,
    float* __restrict__ ws) {
  const int r = blockIdx.x;  // 0..31 : 0..15 -> A2, 16..31 -> A3
  const float* A = (r < 16) ? (A2 + (size_t)r * H_DIM)
                            : (A3 + (size_t)(r - 16) * H_DIM);
  float* W = ws + ((r < 16) ? (WS_W2 + (size_t)r * H_DIM)
                            : (WS_W3 + (size_t)(r - 16) * H_DIM));
  const int t = threadIdx.x;
  float c = 0.f, d = 0.f;
  for (int h = t; h < H_DIM; h += 256) {
    float a = A[h];
    W[h] = lw[h] * a;
    c += lw[h] * a;
    d += lb[h] * a;
  }
  __shared__ float rc[256], rd[256];
  rc[t] = c; rd[t] = d;
  __syncthreads();
  for (int s = 128; s > 0; s >>= 1) {
    if (t < s) { rc[t] += rc[t + s]; rd[t] += rd[t + s]; }
    __syncthreads();
  }
  if (t == 0) {
    int base = (r < 16) ? 0 : 32;
    int rr   = (r < 16) ? r : (r - 16);
    ws[WS_C2 + base + rr]      = rc[0];   // c
    ws[WS_C2 + base + 16 + rr] = rd[0];   // d
  }
}

// ------------------------------------------------------------------
// K1: fused LN1 + K/V projection + partial ss, single pass over H_in.
// grid = 128 (token, L-half), block = 256 (8 waves).
// Each wave: 2 row-tiles x (K,V) = 4 accumulators; row sum/sumsq
// computed from the A fragments inside the WMMA loop (no stats pass).
// ------------------------------------------------------------------
__global__ __launch_bounds__(256) void stage1_kernel(
    const float* __restrict__ Hin, float* __restrict__ ws) {
  __shared__ float cdL[64];          // c2[16] d2[16] c3[16] d3[16]
  __shared__ float Kt[8][16][16];
  __shared__ float Vt[8][16][16];
  __shared__ float ssAcc[256];

  const int bs   = blockIdx.x >> 1;
  const int half = blockIdx.x & 1;
  const int t    = threadIdx.x;
  const int lane = t & 31;
  const int wave = t >> 5;
  const float* X = Hin + (size_t)bs * LSEQ * H_DIM + (size_t)half * 256 * H_DIM;

  if (t < 64) cdL[t] = ws[WS_C2 + t];
  ssAcc[t] = 0.f;
  __syncthreads();

  const int m16 = lane & 15;            // A row / B row-of-transpose
  const int kh  = (lane >> 4) << 1;     // 0 or 2 : K-pair select
  const float* BK = ws + WS_W2 + (size_t)m16 * H_DIM + kh;
  const float* BV = ws + WS_W3 + (size_t)m16 * H_DIM + kh;
  const float* A0 = X + (size_t)(wave * 16 + m16) * H_DIM + kh;        // tile wave
  const float* A1 = X + (size_t)((wave + 8) * 16 + m16) * H_DIM + kh;  // tile wave+8

  const v8f vz = {0.f,0.f,0.f,0.f,0.f,0.f,0.f,0.f};
  v8f accK[2] = {vz, vz}, accV[2] = {vz, vz};
  float s0 = 0.f, q0 = 0.f, s1 = 0.f, q1 = 0.f;

#pragma unroll 2
  for (int k0 = 0; k0 < H_DIM; k0 += 4) {
    v2f bk = *reinterpret_cast<const v2f*>(BK + k0);
    v2f bv = *reinterpret_cast<const v2f*>(BV + k0);
    v2f a0 = *reinterpret_cast<const v2f*>(A0 + k0);
    v2f a1 = *reinterpret_cast<const v2f*>(A1 + k0);
    accK[0] = wmma4(a0, bk, accK[0]);
    accV[0] = wmma4(a0, bv, accV[0]);
    accK[1] = wmma4(a1, bk, accK[1]);
    accV[1] = wmma4(a1, bv, accV[1]);
    s0 += a0.x + a0.y;  q0 = fmaf(a0.x, a0.x, fmaf(a0.y, a0.y, q0));
    s1 += a1.x + a1.y;  q1 = fmaf(a1.x, a1.x, fmaf(a1.y, a1.y, q1));
  }

  // lane m holds (k%4<2)-half of row m16; lane m+16 holds the other half
  float mu[2], rs[2];
  {
    float fs0 = s0 + __shfl_xor(s0, 16, 32);
    float fq0 = q0 + __shfl_xor(q0, 16, 32);
    float fs1 = s1 + __shfl_xor(s1, 16, 32);
    float fq1 = q1 + __shfl_xor(q1, 16, 32);
    mu[0] = fs0 / (float)H_DIM;
    rs[0] = rsqrtf(fq0 / (float)H_DIM - mu[0] * mu[0] + EPS);
    mu[1] = fs1 / (float)H_DIM;
    rs[1] = rsqrtf(fq1 / (float)H_DIM - mu[1] * mu[1] + EPS);
  }

  const int rcol  = lane & 15;          // output column r
  const int mbase = (lane >> 4) << 3;   // 0 or 8 : D-row half
  const float c2 = cdL[rcol],      d2 = cdL[16 + rcol];
  const float c3 = cdL[32 + rcol], d3 = cdL[48 + rcol];

  for (int round = 0; round < 2; ++round) {
#pragma unroll
    for (int i = 0; i < 8; ++i) {
      int ml = i + mbase;                       // row within tile
      float mui = __shfl(mu[round], ml, 32);    // stats live in lane ml
      float rsi = __shfl(rs[round], ml, 32);
      Kt[wave][ml][rcol] = rsi * (accK[round][i] - mui * c2) + d2;
      Vt[wave][ml][rcol] = rsi * (accV[round][i] - mui * c3) + d3;
    }
    __syncthreads();
    int kk = t >> 4, vv = t & 15;
    float a = 0.f;
#pragma unroll
    for (int w = 0; w < 8; ++w)
#pragma unroll
      for (int i = 0; i < 16; ++i)
        a += Kt[w][i][kk] * Vt[w][i][vv];
    ssAcc[t] += a;
    __syncthreads();
  }
  ws[WS_SS + (size_t)(bs * 2 + half) * 256 + t] = ssAcc[t];
}

// ------------------------------------------------------------------
// K2: h_pooled = (ss half0 + half1) @ O_w^T, then LN2. grid = 64.
// ------------------------------------------------------------------
__global__ __launch_bounds__(256) void stage2_kernel(
    const float* __restrict__ Ow, const float* __restrict__ l2w,
    const float* __restrict__ l2b, float* __restrict__ ws) {
  __shared__ float ssL[256];
  __shared__ float red[256];
  const int tok = blockIdx.x, t = threadIdx.x;
  ssL[t] = ws[WS_SS + (size_t)tok * 512 + t] +
           ws[WS_SS + (size_t)tok * 512 + 256 + t];
  __syncthreads();
  float hp[8];
#pragma unroll
  for (int j = 0; j < 8; ++j) {
    int h = t + 256 * j;
    const float4* Orow = reinterpret_cast<const float4*>(Ow + (size_t)h * 256);
    float acc = 0.f;
#pragma unroll 4
    for (int k = 0; k < 64; ++k) {
      float4 o = Orow[k];
      const float* s4 = &ssL[k * 4];
      acc += o.x * s4[0] + o.y * s4[1] + o.z * s4[2] + o.w * s4[3];
    }
    hp[j] = acc;
    ws[WS_HP + (size_t)tok * H_DIM + h] = acc;
  }
  float s = 0.f;
#pragma unroll
  for (int j = 0; j < 8; ++j) s += hp[j];
  red[t] = s; __syncthreads();
  for (int st = 128; st > 0; st >>= 1) {
    if (t < st) red[t] += red[t + st];
    __syncthreads();
  }
  float mu = red[0] / (float)H_DIM;
  __syncthreads();
  float sq = 0.f;
#pragma unroll
  for (int j = 0; j < 8; ++j) { float d = hp[j] - mu; sq += d * d; }
  red[t] = sq; __syncthreads();
  for (int st = 128; st > 0; st >>= 1) {
    if (t < st) red[t] += red[t + st];
    __syncthreads();
  }
  float rsg = rsqrtf(red[0] / (float)H_DIM + EPS);
#pragma unroll
  for (int j = 0; j < 8; ++j) {
    int h = t + 256 * j;
    ws[WS_HN + (size_t)tok * H_DIM + h] = (hp[j] - mu) * rsg * l2w[h] + l2b[h];
  }
}

// ------------------------------------------------------------------
// K3: ht = gelu(hn @ ffn_w1^T + b1). Each wave: one N-tile shared
// across 2 M-tiles (B loaded once per 2 WMMAs). 1024 waves.
// ------------------------------------------------------------------
__global__ __launch_bounds__(256) void ffn1_kernel(
    const float* __restrict__ w1, const float* __restrict__ b1,
    float* __restrict__ ws) {
  const int t = threadIdx.x, lane = t & 31, wave = t >> 5;
  const int wid = blockIdx.x * 8 + wave;   // 0..1023
  const int mp  = wid >> 9;                // 0..1  (M-tile pair)
  const int nt  = wid & 511;               // 0..511
  const int m16 = lane & 15;
  const int kh  = (lane >> 4) << 1;
  const float* B  = w1 + (size_t)(nt * 16 + m16) * H_DIM + kh;
  const float* A0 = ws + WS_HN + (size_t)(mp * 32 + m16) * H_DIM + kh;
  const v8f vz = {0.f,0.f,0.f,0.f,0.f,0.f,0.f,0.f};
  v8f acc[2] = {vz, vz};
#pragma unroll 2
  for (int k0 = 0; k0 < H_DIM; k0 += 4) {
    v2f b = *reinterpret_cast<const v2f*>(B + k0);
#pragma unroll
    for (int m = 0; m < 2; ++m) {
      v2f a = *reinterpret_cast<const v2f*>(A0 + (size_t)m * 16 * H_DIM + k0);
      acc[m] = wmma4(a, b, acc[m]);
    }
  }
  const int n = nt * 16 + (lane & 15);
  const float bias = b1[n];
  const int mbase = (lane >> 4) << 3;
#pragma unroll
  for (int m = 0; m < 2; ++m)
#pragma unroll
    for (int i = 0; i < 8; ++i) {
      int tok = mp * 32 + m * 16 + i + mbase;
      float x = acc[m][i] + bias;
      float g = 0.5f * x * (1.0f + erff(x * 0.70710678118654752f));
      ws[WS_HT + (size_t)tok * FFN + n] = g;
    }
}

// ------------------------------------------------------------------
// K4: split-K=2 partials of ht @ ffn_w2^T. Each wave: 2 M-tiles per
// N-tile per K-half. 512 waves (64 blocks).
// ------------------------------------------------------------------
__global__ __launch_bounds__(256) void ffn2_kernel(
    const float* __restrict__ w2, float* __restrict__ ws) {
  const int t = threadIdx.x, lane = t & 31, wave = t >> 5;
  const int wid = blockIdx.x * 8 + wave;   // 0..511
  const int ks  = wid >> 8;                // 0..1   K-split half
  const int rem = wid & 255;
  const int mp  = rem >> 7;                // 0..1   M-tile pair
  const int nt  = rem & 127;               // 0..127
  const int m16 = lane & 15;
  const int kh  = (lane >> 4) << 1;
  const int kofs = ks * (FFN / 2);
  const float* B  = w2 + (size_t)(nt * 16 + m16) * FFN + kofs + kh;
  const float* A0 = ws + WS_HT + (size_t)(mp * 32 + m16) * FFN + kofs + kh;
  const v8f vz = {0.f,0.f,0.f,0.f,0.f,0.f,0.f,0.f};
  v8f acc[2] = {vz, vz};
#pragma unroll 2
  for (int k0 = 0; k0 < FFN / 2; k0 += 4) {
    v2f b = *reinterpret_cast<const v2f*>(B + k0);
#pragma unroll
    for (int m = 0; m < 2; ++m) {
      v2f a = *reinterpret_cast<const v2f*>(A0 + (size_t)m * 16 * FFN + k0);
      acc[m] = wmma4(a, b, acc[m]);
    }
  }
  const int n = nt * 16 + (lane & 15);
  const int mbase = (lane >> 4) << 3;
#pragma unroll
  for (int m = 0; m < 2; ++m)
#pragma unroll
    for (int i = 0; i < 8; ++i) {
      int tok = mp * 32 + m * 16 + i + mbase;
      ws[WS_P2 + (size_t)ks * (64 * H_DIM) + (size_t)tok * H_DIM + n] = acc[m][i];
    }
}

// ------------------------------------------------------------------
// K5: out = h_pooled + partial0 + partial1 + b2. 131072 elements.
// ------------------------------------------------------------------
__global__ __launch_bounds__(256) void combine_kernel(
    const float* __restrict__ b2, const float* __restrict__ ws,
    float* __restrict__ out) {
  const int idx = blockIdx.x * 256 + threadIdx.x;   // 0..131071
  out[idx] = ws[WS_HP + idx] + ws[WS_P2 + idx] +
             ws[WS_P2 + 64 * H_DIM + idx] + b2[idx & (H_DIM - 1)];
}

extern "C" void kernel_launch(void* const* d_in, const int* in_sizes, int n_in,
                              void* d_out, int out_size, void* d_ws, size_t ws_size,
                              hipStream_t stream) {
  const float* Hin = (const float*)d_in[0];
  const float* A2  = (const float*)d_in[1];
  const float* A3  = (const float*)d_in[2];
  const float* Ow  = (const float*)d_in[3];
  const float* w1  = (const float*)d_in[4];
  const float* b1  = (const float*)d_in[5];
  const float* w2  = (const float*)d_in[6];
  const float* b2  = (const float*)d_in[7];
  const float* l1w = (const float*)d_in[8];
  const float* l1b = (const float*)d_in[9];
  const float* l2w = (const float*)d_in[10];
  const float* l2b = (const float*)d_in[11];
  float* ws  = (float*)d_ws;
  float* out = (float*)d_out;

  hipLaunchKernelGGL(prep_kernel,    dim3(32),  dim3(256), 0, stream, A2, A3, l1w, l1b, ws);
  hipLaunchKernelGGL(stage1_kernel,  dim3(128), dim3(256), 0, stream, Hin, ws);
  hipLaunchKernelGGL(stage2_kernel,  dim3(64),  dim3(256), 0, stream, Ow, l2w, l2b, ws);
  hipLaunchKernelGGL(ffn1_kernel,    dim3(128), dim3(256), 0, stream, w1, b1, ws);
  hipLaunchKernelGGL(ffn2_kernel,    dim3(64),  dim3(256), 0, stream, w2, ws);
  hipLaunchKernelGGL(combine_kernel, dim3(512), dim3(256), 0, stream, b2, ws, out);
}